// CatLayer_80934363726437
// MI455X (gfx1250) — compile-verified
//
#include <hip/hip_runtime.h>
#include <hip/hip_bf16.h>

// CatLayer on MI455X (gfx1250):
//   out[i,j,b,c] = A[i,b,c] + B[j,b,c]
//   A = leaky(x) @ W[:, :512]^T + bias   (1536x512, K=512)
//   B = leaky(x) @ W[:, 512:]^T          (1536x512, K=512)
// Stage 1: fp32 WMMA (v_wmma_f32_16x16x4_f32) GEMM -> d_ws (6 MB).
// Stage 2: float4 streaming broadcast-add -> 302 MB output. Output stores are
//          non-temporal so the write stream does not evict the L2-resident
//          A/B tiles (6 MB vs 192 MB L2); HBM-write bound at 23.3 TB/s.

#define L 96
#define BATCH 16
#define D 512
#define M_ROWS (L * BATCH)        // 1536
#define AB_ELEMS (M_ROWS * D)     // 786432 floats per half

typedef float v2f __attribute__((ext_vector_type(2)));
typedef float v4f __attribute__((ext_vector_type(4)));
typedef float v8f __attribute__((ext_vector_type(8)));

__device__ __forceinline__ float leaky(float v) {
    return v > 0.0f ? v : v * 0.1f;
}

// One wave per 16x64 strip of the combined (1536 x 1024) result.
// grid = (96, 16): blockIdx.x -> M tile (16 rows), blockIdx.y -> 64-col quad.
// Quads 0..7 are the A half (W[:, :512], + bias), quads 8..15 the B half.
__global__ __launch_bounds__(32)
void catlayer_gemm_wmma(const float* __restrict__ x,     // (1536, 512) row-major
                        const float* __restrict__ W,     // (512, 1024) row-major
                        const float* __restrict__ bias,  // (512)
                        float* __restrict__ AB)          // (2, 1536, 512)
{
    const int lane = threadIdx.x;       // 0..31
    const int half = lane >> 4;         // 0 or 1 (K sub-group per ISA layout)
    const int lr   = lane & 15;

    const int  m0    = blockIdx.x * 16;       // output row base
    const int  n0    = blockIdx.y * 64;       // combined col base (0..1023)
    const bool isA   = (n0 < D);
    const int  ccol  = n0 & (D - 1);          // output channel base (0..511)
    const int  kbase = isA ? 0 : D;           // K offset into W's 1024 columns

    v8f acc0 = {}, acc1 = {}, acc2 = {}, acc3 = {};

    // A fragment rows: lane lr owns matrix row m0+lr (both halves).
    const float* xrow = x + (m0 + lr) * D;
    // B fragment rows: tile t, lane lr owns output channel ccol + 16*t + lr.
    const float* wr0 = W + (ccol +  0 + lr) * (2 * D) + kbase;
    const float* wr1 = W + (ccol + 16 + lr) * (2 * D) + kbase;
    const float* wr2 = W + (ccol + 32 + lr) * (2 * D) + kbase;
    const float* wr3 = W + (ccol + 48 + lr) * (2 * D) + kbase;

    #pragma unroll 4
    for (int k = 0; k < D; k += 4) {
        const int ka = k + 2 * half;    // this lane's K pair: {ka, ka+1}

        // A fragment (16x4 f32): VGPR0 = K=2*half, VGPR1 = K=2*half+1
        v2f a;
        a.x = leaky(xrow[ka]);
        a.y = leaky(xrow[ka + 1]);

        // B fragments (4x16 f32), mirrored layout: B[k,n] = W[n, kbase+k]
        v2f b0 = { wr0[ka], wr0[ka + 1] };
        v2f b1 = { wr1[ka], wr1[ka + 1] };
        v2f b2 = { wr2[ka], wr2[ka + 1] };
        v2f b3 = { wr3[ka], wr3[ka + 1] };

        acc0 = __builtin_amdgcn_wmma_f32_16x16x4_f32(false, a, false, b0,
                                                     (short)0, acc0, false, false);
        acc1 = __builtin_amdgcn_wmma_f32_16x16x4_f32(false, a, false, b1,
                                                     (short)0, acc1, false, false);
        acc2 = __builtin_amdgcn_wmma_f32_16x16x4_f32(false, a, false, b2,
                                                     (short)0, acc2, false, false);
        acc3 = __builtin_amdgcn_wmma_f32_16x16x4_f32(false, a, false, b3,
                                                     (short)0, acc3, false, false);
    }

    // Fold bias into the A half only (added once per output element).
    if (isA) {
        const float bv0 = bias[ccol +  0 + lr];
        const float bv1 = bias[ccol + 16 + lr];
        const float bv2 = bias[ccol + 32 + lr];
        const float bv3 = bias[ccol + 48 + lr];
        #pragma unroll
        for (int r = 0; r < 8; ++r) {
            acc0[r] += bv0; acc1[r] += bv1; acc2[r] += bv2; acc3[r] += bv3;
        }
    }

    // C/D layout: lane -> N = lr, VGPR r -> M = r + 8*half.
    float* dst = AB + (isA ? 0 : AB_ELEMS);
    #pragma unroll
    for (int r = 0; r < 8; ++r) {
        const int m = m0 + r + 8 * half;
        float* row = dst + m * D + ccol + lr;
        row[ 0] = acc0[r];
        row[16] = acc1[r];
        row[32] = acc2[r];
        row[48] = acc3[r];
    }
}

// out[((i*L + j)*BATCH + bb)*D + c] = A[(i*BATCH+bb)*D + c] + B[(j*BATCH+bb)*D + c]
// 128-bit vectorized; A/B (3 MB each) stay L2-resident; write stream = 302 MB
// pushed out with non-temporal stores so it does not thrash L2.
__global__ __launch_bounds__(256)
void catlayer_combine(const v4f* __restrict__ A4,
                      const v4f* __restrict__ B4,
                      v4f* __restrict__ out4)
{
    const int idx = blockIdx.x * 256 + threadIdx.x;   // one v4f each
    const int c4   = idx & (D / 4 - 1);               // 0..127
    const int rest = idx >> 7;                        // (i*L + j)*BATCH + bb
    const int bb   = rest & (BATCH - 1);
    const int ij   = rest >> 4;
    const int i    = ij / L;
    const int j    = ij - i * L;

    const v4f a = A4[(i * BATCH + bb) * (D / 4) + c4];
    const v4f b = B4[(j * BATCH + bb) * (D / 4) + c4];
    const v4f r = a + b;
    __builtin_nontemporal_store(r, &out4[idx]);       // th:NT write stream
}

extern "C" void kernel_launch(void* const* d_in, const int* in_sizes, int n_in,
                              void* d_out, int out_size, void* d_ws, size_t ws_size,
                              hipStream_t stream) {
    const float* x    = (const float*)d_in[0];   // (96, 16, 512)
    const float* W    = (const float*)d_in[1];   // (512, 1024)
    const float* bias = (const float*)d_in[2];   // (512)
    float*       out  = (float*)d_out;           // (9216, 16, 512)
    float*       AB   = (float*)d_ws;            // 2 * 1536 * 512 floats = 6 MB

    // Stage 1: WMMA GEMM for both halves (A gets bias folded in).
    dim3 g1(M_ROWS / 16, (2 * D) / 64);          // (96, 16) waves of 32
    catlayer_gemm_wmma<<<g1, 32, 0, stream>>>(x, W, bias, AB);

    // Stage 2: streaming broadcast-add with NT stores.
    const int total4 = L * L * BATCH * (D / 4);  // 18,874,368 v4f (exact)
    catlayer_combine<<<total4 / 256, 256, 0, stream>>>(
        (const v4f*)AB, (const v4f*)(AB + AB_ELEMS), (v4f*)out);
}